// JointsAcc_16930761080839
// MI455X (gfx1250) — compile-verified
//
#include <hip/hip_runtime.h>
#include <float.h>
#include <math.h>

#define B_  128
#define J_  17
#define H_  128
#define W_  128
#define HM_SIZE (H_ * W_)      // 16384 elements per heatmap
#define NHM (B_ * J_)          // 2176 heatmaps per tensor
#define NT  256                // 8 wave32 waves per block

// ---------------------------------------------------------------------------
// Kernel A: per-heatmap argmax.  blockIdx.x in [0, 2*NHM): first half works on
// `output`, second half on `target`.  Each block streams one 64 KB heatmap with
// coalesced float4 (global_load_b128) loads, tracks (maxval, first-index), then
// reduces with a wave32 shfl_xor butterfly + 8-entry LDS combine.
// ---------------------------------------------------------------------------
__global__ __launch_bounds__(NT) void hm_argmax_kernel(
    const float* __restrict__ out_hm,
    const float* __restrict__ tgt_hm,
    float* __restrict__ pred_out,   // [NHM][2] -> final location in d_out
    float* __restrict__ targ_out)   // [NHM][2] -> workspace
{
    const int  bid    = blockIdx.x;
    const bool is_tgt = bid >= NHM;
    const int  hm     = is_tgt ? (bid - NHM) : bid;
    const float4* __restrict__ b4 =
        (const float4*)((is_tgt ? tgt_hm : out_hm) + (size_t)hm * HM_SIZE);

    const int t = threadIdx.x;
    float best    = -FLT_MAX;
    int   bestIdx = 0x7FFFFFFF;

    // 4096 float4 chunks; thread t covers chunks t, t+256, ... (16 iters).
    // Unroll 4 keeps several B128 loads in flight without blowing VGPRs.
    #pragma unroll 4
    for (int i = 0; i < 16; ++i) {
        const int q = t + i * NT;
        // gfx1250 global_prefetch_b8: pull the back half of the heatmap into
        // cache while the front half streams (in-bounds: q+2048 <= 4095).
        if (i < 8) __builtin_prefetch((const void*)(b4 + q + 2048), 0, 0);
        const float4 v = b4[q];
        const int e = q << 2;
        if (v.x > best) { best = v.x; bestIdx = e;     }
        if (v.y > best) { best = v.y; bestIdx = e + 1; }
        if (v.z > best) { best = v.z; bestIdx = e + 2; }
        if (v.w > best) { best = v.w; bestIdx = e + 3; }
    }

    // wave32 butterfly; prefer larger value, tie-break on smaller index
    // (matches jnp.argmax first-occurrence semantics).
    #pragma unroll
    for (int off = 16; off > 0; off >>= 1) {
        const float ov = __shfl_xor(best, off, 32);
        const int   oi = __shfl_xor(bestIdx, off, 32);
        if (ov > best || (ov == best && oi < bestIdx)) { best = ov; bestIdx = oi; }
    }

    __shared__ float sval[NT / 32];
    __shared__ int   sidx[NT / 32];
    const int wave = t >> 5;
    const int lane = t & 31;
    if (lane == 0) { sval[wave] = best; sidx[wave] = bestIdx; }
    __syncthreads();

    if (wave == 0) {
        best    = (lane < NT / 32) ? sval[lane] : -FLT_MAX;
        bestIdx = (lane < NT / 32) ? sidx[lane] : 0x7FFFFFFF;
        #pragma unroll
        for (int off = NT / 64; off > 0; off >>= 1) {   // 4,2,1 over 8 waves
            const float ov = __shfl_xor(best, off, 32);
            const int   oi = __shfl_xor(bestIdx, off, 32);
            if (ov > best || (ov == best && oi < bestIdx)) { best = ov; bestIdx = oi; }
        }
        if (lane == 0) {
            const float m = (best > 0.0f) ? 1.0f : 0.0f;  // preds * (maxval>0)
            float2 p;
            p.x = (float)(bestIdx % W_) * m;
            p.y = (float)(bestIdx / W_) * m;
            ((float2*)(is_tgt ? targ_out : pred_out))[hm] = p;
        }
    }
}

// ---------------------------------------------------------------------------
// Kernel B: tiny epilogue (single block).  Per-joint hit/valid counts via LDS
// atomics, then acc[17], avg_acc, cnt with exact reference semantics.
// ---------------------------------------------------------------------------
__global__ __launch_bounds__(NT) void acc_kernel(
    const float* __restrict__ pred,   // d_out + 19
    const float* __restrict__ targ,   // workspace
    float* __restrict__ out)          // d_out: acc[17] | avg | cnt | ...
{
    __shared__ int   s_num[J_];
    __shared__ int   s_hit[J_];
    __shared__ float s_acc[J_];

    const int t = threadIdx.x;
    if (t < J_) { s_num[t] = 0; s_hit[t] = 0; }
    __syncthreads();

    const float normx = (float)H_ / 10.0f;   // 12.8f, rounded exactly as jnp
    const float normy = (float)W_ / 10.0f;

    for (int p = t; p < NHM; p += NT) {
        const int j = p % J_;                         // p = b*J_ + j
        const float px = pred[2 * p];
        const float py = pred[2 * p + 1];
        const float tx = targ[2 * p];
        const float ty = targ[2 * p + 1];
        if (tx > 1.0f && ty > 1.0f) {                 // valid -> dists != -1
            const float dx = (px - tx) / normx;       // IEEE div, match XLA
            const float dy = (py - ty) / normy;
            const float d  = sqrtf(dx * dx + dy * dy);
            atomicAdd(&s_num[j], 1);
            if (d < 0.5f) atomicAdd(&s_hit[j], 1);    // DIST_THRES
        }
    }
    __syncthreads();

    if (t < J_) {
        const int n = s_num[t];
        const float a = (n > 0) ? ((float)s_hit[t] / (float)n) : -1.0f;
        s_acc[t] = a;
        out[t]   = a;
    }
    __syncthreads();

    if (t == 0) {
        float sum = 0.0f;
        int   cnt = 0;
        for (int j = 0; j < J_; ++j)
            if (s_acc[j] >= 0.0f) { sum += s_acc[j]; ++cnt; }
        out[J_]     = (cnt > 0) ? (sum / (float)cnt) : 0.0f;  // avg_acc
        out[J_ + 1] = (float)cnt;                             // cnt
    }
}

// ---------------------------------------------------------------------------
extern "C" void kernel_launch(void* const* d_in, const int* in_sizes, int n_in,
                              void* d_out, int out_size, void* d_ws, size_t ws_size,
                              hipStream_t stream) {
    (void)in_sizes; (void)n_in; (void)out_size; (void)ws_size;
    const float* out_hm = (const float*)d_in[0];   // output [B,J,H,W] f32
    const float* tgt_hm = (const float*)d_in[1];   // target [B,J,H,W] f32
    float* out      = (float*)d_out;
    float* pred_out = out + (J_ + 2);              // pred lands at offset 19
    float* targ_ws  = (float*)d_ws;                // NHM*2 floats of scratch

    hm_argmax_kernel<<<2 * NHM, NT, 0, stream>>>(out_hm, tgt_hm, pred_out, targ_ws);
    acc_kernel<<<1, NT, 0, stream>>>(pred_out, targ_ws, out);
}